// PermutationalLayer_16655883174311
// MI455X (gfx1250) — compile-verified
//
#include <hip/hip_runtime.h>
#include <stdint.h>

// Problem constants (reference: DATA_DIM=4096, BATCH=16384, fp32, int32 idx)
#define D               4096
#define TPB             256
#define ROWS_PER_BLOCK  8
#define CHUNKS          (D / (TPB * 4))       // float4 chunks per thread per row = 4
#define COPY_ROUNDS     (D * 4 / (TPB * 16))  // 16B per lane per round = 4

// Native vector type accepted by __builtin_nontemporal_store (HIP float4 is a class).
typedef float v4f __attribute__((ext_vector_type(4)));

// Per-lane async DMA: 16 bytes global -> LDS, tracked by ASYNCcnt.
// VDST (%0) = LDS byte address (relative to workgroup LDS base),
// VADDR (%1) = 64-bit global address, no SADDR ("off").
__device__ __forceinline__ void async_copy_b128(uint32_t lds_byte_off, const float* gaddr) {
    asm volatile("global_load_async_to_lds_b128 %0, %1, off"
                 :: "v"(lds_byte_off), "v"(gaddr)
                 : "memory");
}

__device__ __forceinline__ void wait_asynccnt_le4() {
    asm volatile("s_wait_asynccnt 4" ::: "memory");
}
__device__ __forceinline__ void wait_asynccnt_0() {
    asm volatile("s_wait_asynccnt 0" ::: "memory");
}

__global__ void __launch_bounds__(TPB)
perm_gather_kernel(const float* __restrict__ z,
                   const int*   __restrict__ idx,
                   float*       __restrict__ out,
                   int nrows) {
    __shared__ float lbuf[2][D];   // 32 KB double buffer

    const int t = threadIdx.x;

    // Cache this thread's gather indices once (shared by every row).
    int cidx[CHUNKS][4];
#pragma unroll
    for (int c = 0; c < CHUNKS; ++c) {
#pragma unroll
        for (int j = 0; j < 4; ++j) {
            cidx[c][j] = idx[c * (TPB * 4) + t * 4 + j];
        }
    }

    // LDS byte offsets (generic shared pointer low 32 bits == LDS offset).
    const uint32_t lds0 = (uint32_t)(uintptr_t)(void*)&lbuf[0][0];
    const uint32_t lds1 = (uint32_t)(uintptr_t)(void*)&lbuf[1][0];

    const int row0 = blockIdx.x * ROWS_PER_BLOCK;
    if (row0 >= nrows) return;
    const int rows_here = (nrows - row0 < ROWS_PER_BLOCK) ? (nrows - row0) : ROWS_PER_BLOCK;

    // Prologue: kick off row 0 into buffer 0.
    {
        const float* src = z + (size_t)row0 * D;
#pragma unroll
        for (int p = 0; p < COPY_ROUNDS; ++p) {
            const int e = p * TPB + t;              // 16B slice index within the row
            async_copy_b128(lds0 + (uint32_t)e * 16u, src + (size_t)e * 4);
        }
    }

    for (int rr = 0; rr < rows_here; ++rr) {
        const int cur = rr & 1;
        const bool have_next = (rr + 1) < rows_here;

        if (have_next) {
            // Issue next row into the other buffer (last gathered in iteration rr-1;
            // sealed by the trailing barrier of iteration rr-1).
            const float* src = z + (size_t)(row0 + rr + 1) * D;
            const uint32_t lb = cur ? lds0 : lds1;
#pragma unroll
            for (int p = 0; p < COPY_ROUNDS; ++p) {
                const int e = p * TPB + t;
                async_copy_b128(lb + (uint32_t)e * 16u, src + (size_t)e * 4);
            }
            // 4 newest (row rr+1) may stay outstanding; the 4 oldest (row rr)
            // must have completed (async loads complete in order per wave).
            wait_asynccnt_le4();
        } else {
            wait_asynccnt_0();
        }
        __syncthreads();   // publish every wave's LDS slice of row rr

        // Gather from LDS, store coalesced 16B per lane, streaming (NT).
        const float* buf = lbuf[cur];
        float* dst = out + (size_t)(row0 + rr) * D;
#pragma unroll
        for (int c = 0; c < CHUNKS; ++c) {
            v4f v;
            v.x = buf[cidx[c][0]];
            v.y = buf[cidx[c][1]];
            v.z = buf[cidx[c][2]];
            v.w = buf[cidx[c][3]];
            __builtin_nontemporal_store(
                v, reinterpret_cast<v4f*>(dst + c * (TPB * 4) + t * 4));
        }
        __syncthreads();   // protect buf[cur] before it is re-filled at rr+2
    }
}

extern "C" void kernel_launch(void* const* d_in, const int* in_sizes, int n_in,
                              void* d_out, int out_size, void* d_ws, size_t ws_size,
                              hipStream_t stream) {
    const float* z   = (const float*)d_in[0];
    const int*   idx = (const int*)d_in[1];
    float*       out = (float*)d_out;

    const int nrows  = in_sizes[0] / D;           // 16384
    const int blocks = (nrows + ROWS_PER_BLOCK - 1) / ROWS_PER_BLOCK;  // 2048

    perm_gather_kernel<<<blocks, TPB, 0, stream>>>(z, idx, out, nrows);
}